// EwaldBlock_7198365188503
// MI455X (gfx1250) — compile-verified
//
#include <hip/hip_runtime.h>
#include <hip/hip_bf16.h>

// EwaldBlock for MI455X (gfx1250): all contractions via V_WMMA_F32_16X16X4_F32
// (f32 A/B, f32 accum -> matches f32 reference numerics), LDS-staged operands,
// sorted-batch segments handled as dense per-segment GEMMs (no atomics).
// Round 3: fast-silu via v_rcp_f32 (drops the IEEE div_scale/div_fixup chain
// from every GEMM epilogue); structure otherwise identical to round 2.

#define N_NODES 16384
#define KDIM    64
#define DDIM    128
#define PDIM    8
#define BSEG    32
#define LN_EPS  1e-5f

#define XS 132   // padded LDS row stride for [.][128] f32 (132 % 64 == 4 -> no bank conflicts)
#define RS 68    // padded LDS row stride for [.][64]  f32
#define XQS 36   // padded LDS row stride for [.][32]  f32

#define MROWS  64   // rows per MLP block
#define DSPLIT 4    // scatter: D-dimension split (32 cols per block)
#define NSPLIT 8    // gather: node-tile round-robin split

typedef __attribute__((ext_vector_type(2))) float v2f;
typedef __attribute__((ext_vector_type(8))) float v8f;

// D = A(16x4, f32) * B(4x16, f32) + C(16x16, f32)
__device__ __forceinline__ v8f wmma4(v2f a, v2f b, v8f c) {
  return __builtin_amdgcn_wmma_f32_16x16x4_f32(
      /*neg_a=*/false, a, /*neg_b=*/false, b,
      /*c_mod=*/(short)0, c, /*reuse_a=*/false, /*reuse_b=*/false);
}

// silu via v_exp_f32 + v_rcp_f32 (single-instruction reciprocal, ~1 ULP)
__device__ __forceinline__ float silu_f(float x) {
  float e = __expf(-x);
  return x * __builtin_amdgcn_rcpf(1.0f + e);
}

__device__ __forceinline__ int seg_lower_bound(const int* __restrict__ batch, int n, int val) {
  int lo = 0, hi = n;
  while (lo < hi) {
    int mid = (lo + hi) >> 1;
    if (batch[mid] < val) lo = mid + 1; else hi = mid;
  }
  return lo;
}

// ---------------------------------------------------------------- trig factors
__global__ void trig_kernel(const float* __restrict__ kdr, const float* __restrict__ damp,
                            float* __restrict__ re, float* __restrict__ im, int total) {
  int i = blockIdx.x * blockDim.x + threadIdx.x;
  if (i < total) {
    float d = damp[i];
    float x = kdr[i];
    re[i] = __cosf(x) * d;
    im[i] = __sinf(x) * d;
  }
}

// ------------------------------------------------------------------- k-filter
// kfilter[k,d] = sum_p down_projection[k,p] * W_up[d,p]
__global__ void kfilter_kernel(const float* __restrict__ dp, const float* __restrict__ wup,
                               float* __restrict__ kf) {
  int i = blockIdx.x * blockDim.x + threadIdx.x;
  if (i < KDIM * DDIM) {
    int k = i / DDIM, d = i % DDIM;
    float acc = 0.f;
#pragma unroll
    for (int p = 0; p < PDIM; ++p) acc += dp[k * PDIM + p] * wup[d * PDIM + p];
    kf[i] = acc;
  }
}

// --------------------------------------------------------- MLP (pre / update)
// out = residual(+LN if mode==0) of  silu(silu(X W1^T) W2^T)
// 64 rows per block, 8 waves; wave w owns (m-tile = w&3, n-half = w>>2).
__global__ __launch_bounds__(256) void mlp_kernel(
    const float* __restrict__ X, const float* __restrict__ W1, const float* __restrict__ W2,
    const float* __restrict__ gamma, const float* __restrict__ beta,
    float* __restrict__ out, int mode) {
  __shared__ float lds_a[MROWS * XS];   // x tile, then h1, then residual for LN
  __shared__ float lds_w[128 * XS];     // W1, then W2
  __shared__ float lds_mu[MROWS];
  __shared__ float lds_rs[MROWS];

  const int tid  = threadIdx.x;
  const int wave = tid >> 5;
  const int lane = tid & 31;
  const int half = lane >> 4;       // 0: K pair {0,1}, 1: K pair {2,3}
  const int l16  = lane & 15;
  const int c0   = half * 2;
  const int row0 = blockIdx.x * MROWS;
  const int mt   = wave & 3;        // m-tile (16 rows)
  const int ntb  = (wave >> 2) * 4; // first of this wave's 4 n-tiles

  // stage X tile (64x128) and W1 (128x128), vectorized (XS%4==0 keeps 16B align)
  for (int idx = tid; idx < MROWS * 32; idx += 256) {
    int r = idx >> 5, c4 = idx & 31;
    float4 xv = ((const float4*)(X + (size_t)(row0 + r) * DDIM))[c4];
    *(float4*)&lds_a[r * XS + c4 * 4] = xv;
  }
  for (int idx = tid; idx < 128 * 32; idx += 256) {
    int r = idx >> 5, c4 = idx & 31;
    float4 wv = ((const float4*)(W1 + (size_t)r * DDIM))[c4];
    *(float4*)&lds_w[r * XS + c4 * 4] = wv;
  }
  // warm L2/WGP$ with W2 while GEMM1 runs (global_prefetch_b8)
  __builtin_prefetch((const char*)W2 + (size_t)tid * 256, 0, 3);
  __builtin_prefetch((const char*)W2 + (size_t)tid * 256 + 128, 0, 3);
  __syncthreads();

  const v8f vzero = {};
  v8f acc[4];

  // GEMM1: h1 = X * W1^T
#pragma unroll
  for (int j = 0; j < 4; ++j) acc[j] = vzero;
  {
    const float* arow = &lds_a[(16 * mt + l16) * XS];
    for (int k0 = 0; k0 < DDIM; k0 += 4) {
      v2f a; a.x = arow[k0 + c0]; a.y = arow[k0 + c0 + 1];
#pragma unroll
      for (int j = 0; j < 4; ++j) {
        const float* brow = &lds_w[(16 * (ntb + j) + l16) * XS];
        v2f b; b.x = brow[k0 + c0]; b.y = brow[k0 + c0 + 1];
        acc[j] = wmma4(a, b, acc[j]);
      }
    }
  }
  // silu -> write h1 (each wave writes its own 16 rows x its 64 cols)
#pragma unroll
  for (int j = 0; j < 4; ++j)
#pragma unroll
    for (int r = 0; r < 8; ++r) {
      int m = 16 * mt + r + 8 * half;
      lds_a[m * XS + 16 * (ntb + j) + l16] = silu_f(acc[j][r]);
    }
  __syncthreads();

  // stage W2
  for (int idx = tid; idx < 128 * 32; idx += 256) {
    int r = idx >> 5, c4 = idx & 31;
    float4 wv = ((const float4*)(W2 + (size_t)r * DDIM))[c4];
    *(float4*)&lds_w[r * XS + c4 * 4] = wv;
  }
  __syncthreads();

  // GEMM2: h2 = h1 * W2^T
#pragma unroll
  for (int j = 0; j < 4; ++j) acc[j] = vzero;
  {
    const float* arow = &lds_a[(16 * mt + l16) * XS];
    for (int k0 = 0; k0 < DDIM; k0 += 4) {
      v2f a; a.x = arow[k0 + c0]; a.y = arow[k0 + c0 + 1];
#pragma unroll
      for (int j = 0; j < 4; ++j) {
        const float* brow = &lds_w[(16 * (ntb + j) + l16) * XS];
        v2f b; b.x = brow[k0 + c0]; b.y = brow[k0 + c0 + 1];
        acc[j] = wmma4(a, b, acc[j]);
      }
    }
  }

  if (mode == 0) {
    // residual + LayerNorm
#pragma unroll
    for (int j = 0; j < 4; ++j)
#pragma unroll
      for (int r = 0; r < 8; ++r) {
        int m = 16 * mt + r + 8 * half;
        int c = 16 * (ntb + j) + l16;
        float xv = X[(size_t)(row0 + m) * DDIM + c];
        lds_a[m * XS + c] = xv + silu_f(acc[j][r]);
      }
    __syncthreads();
    if (tid < MROWS) {
      const float* rowp = &lds_a[tid * XS];
      float s = 0.f, s2 = 0.f;
      for (int c = 0; c < DDIM; ++c) { float v = rowp[c]; s += v; s2 += v * v; }
      float mu  = s * (1.f / DDIM);
      float var = s2 * (1.f / DDIM) - mu * mu;
      lds_mu[tid] = mu;
      lds_rs[tid] = rsqrtf(var + LN_EPS);
    }
    __syncthreads();
    for (int idx = tid; idx < MROWS * DDIM; idx += 256) {
      int r = idx >> 7, c = idx & 127;
      float v = (lds_a[r * XS + c] - lds_mu[r]) * lds_rs[r] * gamma[c] + beta[c];
      out[(size_t)(row0 + r) * DDIM + c] = v;
    }
  } else {
    // residual only
#pragma unroll
    for (int j = 0; j < 4; ++j)
#pragma unroll
      for (int r = 0; r < 8; ++r) {
        int m = 16 * mt + r + 8 * half;
        int c = 16 * (ntb + j) + l16;
        size_t g = (size_t)(row0 + m) * DDIM + c;
        out[g] = X[g] + silu_f(acc[j][r]);
      }
  }
}

// --------------------------------------------------- scatter: structure factors
// sf[b,k,d] = kfilter[k,d] * sum_{n in segment b} trig[n,k] * x_res[n,d]
// grid (BSEG, DSPLIT); block (b,q) covers d in [32q, 32q+32).
// wave w: k-tile = w>>1, local d-tile = w&1.
__global__ __launch_bounds__(256) void scatter_kernel(
    const float* __restrict__ re, const float* __restrict__ im,
    const float* __restrict__ xres, const int* __restrict__ batch,
    const float* __restrict__ kf, float* __restrict__ sfR, float* __restrict__ sfI) {
  __shared__ float lds_r[16 * RS];
  __shared__ float lds_i[16 * RS];
  __shared__ float lds_x[16 * XQS];

  const int b    = blockIdx.x;
  const int q    = blockIdx.y;      // d-quarter
  const int tid  = threadIdx.x;
  const int wave = tid >> 5;
  const int lane = tid & 31;
  const int half = lane >> 4, l16 = lane & 15, c0 = half * 2;
  const int start = seg_lower_bound(batch, N_NODES, b);
  const int end   = seg_lower_bound(batch, N_NODES, b + 1);

  const int kt  = wave >> 1;        // 0..3
  const int dtl = wave & 1;         // local d-tile within the 32-wide slice

  v8f accR = {}, accI = {};

  for (int n0 = start; n0 < end; n0 += 16) {
    for (int idx = tid; idx < 16 * 64; idx += 256) {
      int r = idx >> 6, k = idx & 63;
      int n = n0 + r;
      float rv = 0.f, iv = 0.f;
      if (n < end) { rv = re[(size_t)n * KDIM + k]; iv = im[(size_t)n * KDIM + k]; }
      lds_r[r * RS + k] = rv;
      lds_i[r * RS + k] = iv;
    }
    for (int idx = tid; idx < 16 * 32; idx += 256) {
      int r = idx >> 5, c = idx & 31;
      int n = n0 + r;
      lds_x[r * XQS + c] = (n < end) ? xres[(size_t)n * DDIM + q * 32 + c] : 0.f;
    }
    __syncthreads();
#pragma unroll
    for (int cb = 0; cb < 16; cb += 4) {   // contraction over 4 nodes per WMMA
      v2f ar; ar.x = lds_r[(cb + c0) * RS + 16 * kt + l16];
              ar.y = lds_r[(cb + c0 + 1) * RS + 16 * kt + l16];
      v2f ai; ai.x = lds_i[(cb + c0) * RS + 16 * kt + l16];
              ai.y = lds_i[(cb + c0 + 1) * RS + 16 * kt + l16];
      int dl = 16 * dtl + l16;
      v2f bx; bx.x = lds_x[(cb + c0) * XQS + dl];
              bx.y = lds_x[(cb + c0 + 1) * XQS + dl];
      accR = wmma4(ar, bx, accR);
      accI = wmma4(ai, bx, accI);
    }
    __syncthreads();
  }

  // apply k-space filter at store time
#pragma unroll
  for (int r = 0; r < 8; ++r) {
    int k = 16 * kt + r + 8 * half;
    int d = q * 32 + 16 * dtl + l16;
    float f = kf[k * DDIM + d];
    size_t o = ((size_t)b * KDIM + k) * DDIM + d;
    sfR[o] = accR[r] * f;
    sfI[o] = accI[r] * f;
  }
}

// --------------------------------------------------------- gather: ewald message
// x2[n,d] = x_scalar[n,d] + sum_k real[n,k]*sfR[b,k,d] + imag[n,k]*sfI[b,k,d]
// grid (BSEG, NSPLIT); block (b,s) handles node tiles start+16*s, step 16*NSPLIT.
__global__ __launch_bounds__(256) void gather_kernel(
    const float* __restrict__ xsc, const float* __restrict__ re,
    const float* __restrict__ im, const int* __restrict__ batch,
    const float* __restrict__ sfR, const float* __restrict__ sfI,
    float* __restrict__ x2) {
  __shared__ float lds_sr[64 * XS];
  __shared__ float lds_si[64 * XS];
  __shared__ float lds_r[16 * RS];
  __shared__ float lds_i[16 * RS];

  const int b    = blockIdx.x;
  const int s    = blockIdx.y;
  const int tid  = threadIdx.x;
  const int wave = tid >> 5;
  const int lane = tid & 31;
  const int half = lane >> 4, l16 = lane & 15, c0 = half * 2;
  const int start = seg_lower_bound(batch, N_NODES, b);
  const int end   = seg_lower_bound(batch, N_NODES, b + 1);

  for (int idx = tid; idx < 64 * 128; idx += 256) {
    int k = idx >> 7, d = idx & 127;
    size_t o = ((size_t)b * KDIM + k) * DDIM + d;
    lds_sr[k * XS + d] = sfR[o];
    lds_si[k * XS + d] = sfI[o];
  }
  __syncthreads();

  const int dt = wave;               // each wave owns one 16-wide d-tile
  for (int n0 = start + 16 * s; n0 < end; n0 += 16 * NSPLIT) {
    for (int idx = tid; idx < 16 * 64; idx += 256) {
      int r = idx >> 6, k = idx & 63;
      int n = n0 + r;
      lds_r[r * RS + k] = (n < end) ? re[(size_t)n * KDIM + k] : 0.f;
      lds_i[r * RS + k] = (n < end) ? im[(size_t)n * KDIM + k] : 0.f;
    }
    __syncthreads();
    v8f acc = {};
#pragma unroll
    for (int k0 = 0; k0 < KDIM; k0 += 4) {
      v2f ar; ar.x = lds_r[l16 * RS + k0 + c0]; ar.y = lds_r[l16 * RS + k0 + c0 + 1];
      v2f ai; ai.x = lds_i[l16 * RS + k0 + c0]; ai.y = lds_i[l16 * RS + k0 + c0 + 1];
      v2f br; br.x = lds_sr[(k0 + c0) * XS + 16 * dt + l16];
              br.y = lds_sr[(k0 + c0 + 1) * XS + 16 * dt + l16];
      v2f bi; bi.x = lds_si[(k0 + c0) * XS + 16 * dt + l16];
              bi.y = lds_si[(k0 + c0 + 1) * XS + 16 * dt + l16];
      acc = wmma4(ar, br, acc);
      acc = wmma4(ai, bi, acc);
    }
#pragma unroll
    for (int r = 0; r < 8; ++r) {
      int n = n0 + r + 8 * half;
      if (n < end) {
        int d = 16 * dt + l16;
        size_t g = (size_t)n * DDIM + d;
        x2[g] = xsc[g] + acc[r];
      }
    }
    __syncthreads();
  }
}

// ------------------------------------------------------------------- launcher
extern "C" void kernel_launch(void* const* d_in, const int* in_sizes, int n_in,
                              void* d_out, int out_size, void* d_ws, size_t ws_size,
                              hipStream_t stream) {
  const float* x_scalar = (const float*)d_in[0];
  const float* k_dot_r  = (const float*)d_in[1];
  const float* sinc     = (const float*)d_in[2];
  const int*   batch    = (const int*)d_in[3];
  const float* downp    = (const float*)d_in[4];
  const float* W_pre1   = (const float*)d_in[5];
  const float* W_pre2   = (const float*)d_in[6];
  const float* gamma    = (const float*)d_in[7];
  const float* beta     = (const float*)d_in[8];
  const float* W_up     = (const float*)d_in[9];
  const float* W_upd1   = (const float*)d_in[10];
  const float* W_upd2   = (const float*)d_in[11];

  float* ws   = (float*)d_ws;
  float* xres = ws;                                  // N*D
  float* x2   = xres + (size_t)N_NODES * DDIM;       // N*D
  float* rev  = x2   + (size_t)N_NODES * DDIM;       // N*K
  float* imv  = rev  + (size_t)N_NODES * KDIM;       // N*K
  float* sfR  = imv  + (size_t)N_NODES * KDIM;       // B*K*D
  float* sfI  = sfR  + (size_t)BSEG * KDIM * DDIM;   // B*K*D
  float* kf   = sfI  + (size_t)BSEG * KDIM * DDIM;   // K*D

  trig_kernel<<<(N_NODES * KDIM + 255) / 256, 256, 0, stream>>>(
      k_dot_r, sinc, rev, imv, N_NODES * KDIM);
  kfilter_kernel<<<(KDIM * DDIM + 255) / 256, 256, 0, stream>>>(downp, W_up, kf);
  mlp_kernel<<<N_NODES / MROWS, 256, 0, stream>>>(
      x_scalar, W_pre1, W_pre2, gamma, beta, xres, 0);
  scatter_kernel<<<dim3(BSEG, DSPLIT), 256, 0, stream>>>(
      rev, imv, xres, batch, kf, sfR, sfI);
  gather_kernel<<<dim3(BSEG, NSPLIT), 256, 0, stream>>>(
      x_scalar, rev, imv, batch, sfR, sfI, x2);
  mlp_kernel<<<N_NODES / MROWS, 256, 0, stream>>>(
      x2, W_upd1, W_upd2, gamma, beta, (float*)d_out, 1);
}